// topic_model_toy_14164802142733
// MI455X (gfx1250) — compile-verified
//
#include <hip/hip_runtime.h>
#include <hip/hip_bf16.h>

typedef __attribute__((ext_vector_type(2))) float v2f;
typedef __attribute__((ext_vector_type(8))) float v8f;

// D = A(16x4) * B(4x16) + C, all f32.
__device__ __forceinline__ v8f wmma4(v2f a, v2f b, v8f c) {
  return __builtin_amdgcn_wmma_f32_16x16x4_f32(false, a, false, b, (short)0, c, false, false);
}

#define Bsz 128
#define Dd  256
#define Ss  512
#define Kk  50
#define Hh  256
#define KP  64      // K padded to 64
#define NROW 65536  // B*S
#define SPLIT 8     // S-splits per batch row (parallelism)

// ---------------- prep: normalized/paired B panels + concept_far ----------------
// Pair layout: P[(k2*N + n)*2 + j] = M[2*k2+j][n]  -> one b64 load per WMMA B operand.
__global__ __launch_bounds__(256) void prep_kernel(const float* __restrict__ tv,
                                                   const float* __restrict__ W1,
                                                   const float* __restrict__ W2,
                                                   float* __restrict__ tvpP,  // [128][64][2]
                                                   float* __restrict__ W1pP,  // [32][256][2]
                                                   float* __restrict__ W2P,   // [128][256][2]
                                                   float* __restrict__ out) {
  __shared__ float innK[KP];
  __shared__ float red[256];
  int t = threadIdx.x;
  if (t < KP) {
    float s = 0.f;
    if (t < Kk) {
      for (int d = 0; d < Dd; ++d) { float v = tv[d * Kk + t]; s += v * v; }
    }
    innK[t] = (t < Kk) ? 1.f / fmaxf(sqrtf(s), 1e-12f) : 0.f;
  }
  __syncthreads();
  for (int i = t; i < (Dd / 2) * KP; i += 256) {           // tvn paired
    int d2 = i >> 6, n = i & 63;
    float a = (n < Kk) ? tv[(2 * d2) * Kk + n] * innK[n] : 0.f;
    float b = (n < Kk) ? tv[(2 * d2 + 1) * Kk + n] * innK[n] : 0.f;
    tvpP[i * 2] = a; tvpP[i * 2 + 1] = b;
  }
  for (int i = t; i < (KP / 2) * Hh; i += 256) {           // W1 zero-padded + paired
    int k2 = i >> 8, h = i & 255;
    int ka = 2 * k2, kb = 2 * k2 + 1;
    W1pP[i * 2]     = (ka < Kk) ? W1[ka * Hh + h] : 0.f;
    W1pP[i * 2 + 1] = (kb < Kk) ? W1[kb * Hh + h] : 0.f;
  }
  for (int i = t; i < (Hh / 2) * Dd; i += 256) {           // W2 paired
    int h2 = i >> 8, d = i & 255;
    W2P[i * 2]     = W2[(2 * h2) * Dd + d];
    W2P[i * 2 + 1] = W2[(2 * h2 + 1) * Dd + d];
  }
  // concept_far = (sum_{i,j} G_ij - K)/K^2 ; sum G = ||sum_k tvn_k||^2
  {
    float rs = 0.f;
    for (int k = 0; k < Kk; ++k) rs += tv[t * Kk + k] * innK[k];
    red[t] = rs * rs;
  }
  __syncthreads();
  for (int off = 128; off > 0; off >>= 1) {
    if (t < off) red[t] += red[t + off];
    __syncthreads();
  }
  if (t == 0) {
    out[258] = (red[0] - (float)Kk) / ((float)Kk * (float)Kk);
    out[256] = 0.f;
  }
}

// ---------------- main fused kernel: wave g owns (b = g/SPLIT, slice q = g%SPLIT) ----------------
__global__ __launch_bounds__(64) void main_kernel(const float* __restrict__ f_input,
                                                  const float* __restrict__ tvpP,
                                                  const float* __restrict__ W1pP,
                                                  const float* __restrict__ W2P,
                                                  float* __restrict__ out_tpnn,   // [B,S,K]
                                                  float* __restrict__ ws_tpn,     // [K][B*S]
                                                  float* __restrict__ ws_recsum,  // [B*SPLIT][D]
                                                  float* __restrict__ ws_ae) {    // [B*SPLIT]
  __shared__ __align__(16) float xs_s[2][Dd * 16];   // x tile, [d][s]
  __shared__ float nnT_s[2][KP * 16];                // nn transposed [k][m]
  __shared__ float tr_s[2][16 * 16];                 // rec1 chunk transposed [k][m]
  __shared__ float inn_s[2][16];
  __shared__ float rsum_s[2][Dd];

  const int w = threadIdx.x >> 5;
  const int lane = threadIdx.x & 31;
  const int lo = lane & 15, hi = lane >> 4;
  float* xs = xs_s[w];
  float* nnT = nnT_s[w];
  float* tr = tr_s[w];
  float* inn = inn_s[w];
  float* rsum = rsum_s[w];

  const int g = blockIdx.x * 2 + w;       // global wave id
  const int b = g / SPLIT;
  const int q = g % SPLIT;
  const int TPW = (Ss / 16) / SPLIT;      // s-tiles per wave

  for (int i = lane; i < Dd; i += 32) rsum[i] = 0.f;
  float ae_acc = 0.f;

  const float* xb = f_input + (size_t)b * Dd * Ss;

  for (int st = 0; st < TPW; ++st) {
    const int s0 = (q * TPW + st) * 16;
    // ---- load 16x256 x tile into LDS + row sumsq ----
    float ps0 = 0.f, ps1 = 0.f, ps2 = 0.f, ps3 = 0.f;
    const int dbase = lane >> 2;
    const int s4 = (lane & 3) * 4;
    for (int it = 0; it < 32; ++it) {
      int d = it * 8 + dbase;
      const float4 v = *(const float4*)(xb + (size_t)d * Ss + s0 + s4);
      *(float4*)(xs + d * 16 + s4) = v;
      ps0 += v.x * v.x; ps1 += v.y * v.y; ps2 += v.z * v.z; ps3 += v.w * v.w;
    }
    for (int m = 4; m <= 16; m <<= 1) {
      ps0 += __shfl_xor(ps0, m, 32);
      ps1 += __shfl_xor(ps1, m, 32);
      ps2 += __shfl_xor(ps2, m, 32);
      ps3 += __shfl_xor(ps3, m, 32);
    }
    if (lane < 4) {
      inn[lane * 4 + 0] = 1.f / fmaxf(sqrtf(ps0), 1e-12f);
      inn[lane * 4 + 1] = 1.f / fmaxf(sqrtf(ps1), 1e-12f);
      inn[lane * 4 + 2] = 1.f / fmaxf(sqrtf(ps2), 1e-12f);
      inn[lane * 4 + 3] = 1.f / fmaxf(sqrtf(ps3), 1e-12f);
    }

    // ---- GEMM1: topic_prob 16x64 = x(16x256) * tvn(256x64) ----
    v8f acc1[4];
    #pragma unroll
    for (int t = 0; t < 4; ++t) acc1[t] = {};
    for (int kk = 0; kk < Dd / 4; ++kk) {
      int d = kk * 4 + 2 * hi;
      v2f a; a[0] = xs[d * 16 + lo]; a[1] = xs[(d + 1) * 16 + lo];
      #pragma unroll
      for (int t = 0; t < 4; ++t) {
        v2f bv = *(const v2f*)(tvpP + (((size_t)(kk * 2 + hi) * KP + t * 16 + lo) << 1));
        acc1[t] = wmma4(a, bv, acc1[t]);
      }
    }

    // ---- mask + renormalize (topic_prob_n = topic_prob * inv_norm) ----
    v8f nnv[4], tpn[4];
    #pragma unroll
    for (int r = 0; r < 8; ++r) {
      const int row = r + 8 * hi;
      const float iv = inn[row];
      float rsacc = 0.f;
      #pragma unroll
      for (int t = 0; t < 4; ++t) {
        float tp = acc1[t][r];
        float tn = tp * iv;
        tpn[t][r] = tn;
        float am = (tn > 0.3f) ? tp : 0.f;
        nnv[t][r] = am;
        rsacc += am;
      }
      rsacc += __shfl_xor(rsacc, 1, 32);
      rsacc += __shfl_xor(rsacc, 2, 32);
      rsacc += __shfl_xor(rsacc, 4, 32);
      rsacc += __shfl_xor(rsacc, 8, 32);
      float invden = 1.f / (rsacc + 0.001f + 1e-8f);
      #pragma unroll
      for (int t = 0; t < 4; ++t) nnv[t][r] *= invden;
    }

    // ---- emit topic_prob_nn / topic_prob_n, stash nn^T ----
    #pragma unroll
    for (int r = 0; r < 8; ++r) {
      const int row = r + 8 * hi;
      const size_t gr = (size_t)b * Ss + s0 + row;
      #pragma unroll
      for (int t = 0; t < 4; ++t) {
        int col = t * 16 + lo;
        if (col < Kk) {
          out_tpnn[gr * Kk + col] = nnv[t][r];
          ws_tpn[(size_t)col * NROW + gr] = tpn[t][r];
        }
        nnT[col * 16 + row] = nnv[t][r];
      }
    }

    // ---- fused GEMM2(relu) -> GEMM3 ----
    v8f acc3[16];
    #pragma unroll
    for (int nt = 0; nt < 16; ++nt) acc3[nt] = {};
    for (int kc = 0; kc < 4; ++kc) {
      const int c0 = kc * 16;
      v8f acc2 = {};
      for (int kk = 0; kk < KP / 4; ++kk) {
        int k = kk * 4 + 2 * hi;
        v2f a; a[0] = nnT[k * 16 + lo]; a[1] = nnT[(k + 1) * 16 + lo];
        v2f bv = *(const v2f*)(W1pP + (((size_t)(kk * 2 + hi) * Hh + c0 + lo) << 1));
        acc2 = wmma4(a, bv, acc2);
      }
      #pragma unroll
      for (int r = 0; r < 8; ++r)
        tr[lo * 16 + r + 8 * hi] = fmaxf(acc2[r], 0.f);
      for (int kk2 = 0; kk2 < 4; ++kk2) {
        const int kl = kk2 * 4 + 2 * hi;
        v2f a; a[0] = tr[kl * 16 + lo]; a[1] = tr[(kl + 1) * 16 + lo];
        const int kg2 = kc * 8 + kk2 * 2 + hi;
        #pragma unroll
        for (int nt = 0; nt < 16; ++nt) {
          v2f bv = *(const v2f*)(W2P + (((size_t)kg2 * Dd + nt * 16 + lo) << 1));
          acc3[nt] = wmma4(a, bv, acc3[nt]);
        }
      }
    }

    // ---- ae partial + per-column sums ----
    #pragma unroll
    for (int nt = 0; nt < 16; ++nt) {
      const int d = nt * 16 + lo;
      float cs = 0.f;
      #pragma unroll
      for (int r = 0; r < 8; ++r) {
        const int row = r + 8 * hi;
        float rec2 = acc3[nt][r];
        float xn = xs[d * 16 + row] * inn[row];
        float df = xn - rec2;
        ae_acc += df * df;
        cs += rec2;
      }
      cs += __shfl_xor(cs, 16, 32);
      if (hi == 0) rsum[d] += cs;
    }
  }

  for (int i = lane; i < Dd; i += 32) ws_recsum[(size_t)g * Dd + i] = rsum[i];
  for (int m = 16; m >= 1; m >>= 1) ae_acc += __shfl_xor(ae_acc, m, 32);
  if (lane == 0) ws_ae[g] = ae_acc;
}

// ---------------- per-concept top-32: one pass + LDS candidate merge ----------------
__global__ __launch_bounds__(256) void topk_kernel(const float* __restrict__ ws_tpn,
                                                   float* __restrict__ ws_topsum) {
  __shared__ float cand[256 * 32];
  __shared__ float vals[256];
  __shared__ int idxs[256];
  const int k = blockIdx.x;
  const int t = threadIdx.x;
  const float* row = ws_tpn + (size_t)k * NROW;
  float* my = cand + t * 32;

  float mn = 1e30f; int mnp = 0;
  for (int j = 0; j < 32; ++j) {                 // seed with first 32 strided elems
    float v = row[t + j * 256];
    my[j] = v;
    if (v < mn) { mn = v; mnp = j; }
  }
  for (int j = 32; j < 256; ++j) {               // streaming top-32 per thread
    float v = row[t + j * 256];
    if (v > mn) {
      my[mnp] = v;
      mn = my[0]; mnp = 0;
      for (int u = 1; u < 32; ++u) { float c = my[u]; if (c < mn) { mn = c; mnp = u; } }
    }
  }
  __syncthreads();
  float sum = 0.f;
  for (int round = 0; round < Bsz / 4; ++round) {  // merge 8192 candidates -> top-32
    float best = -1e30f; int bi = t * 32;
    for (int u = 0; u < 32; ++u) { float c = my[u]; if (c > best) { best = c; bi = t * 32 + u; } }
    vals[t] = best; idxs[t] = bi;
    __syncthreads();
    for (int off = 128; off > 0; off >>= 1) {
      if (t < off && vals[t + off] > vals[t]) { vals[t] = vals[t + off]; idxs[t] = idxs[t + off]; }
      __syncthreads();
    }
    if (t == 0) { sum += vals[0]; cand[idxs[0]] = -1e30f; }
    __syncthreads();
  }
  if (t == 0) ws_topsum[k] = sum;
}

// ---------------- finalize: pred, ae_loss, concept_sim ----------------
__global__ __launch_bounds__(256) void final_kernel(const float* __restrict__ Wc,
                                                    const float* __restrict__ bcv,
                                                    const float* __restrict__ ws_recsum,
                                                    const float* __restrict__ ws_ae,
                                                    const float* __restrict__ ws_topsum,
                                                    float* __restrict__ out) {
  const int t = threadIdx.x;
  {
    int b = t >> 1, c = t & 1;
    float s = 0.f;
    for (int d = 0; d < Dd; ++d) {
      float rs = 0.f;
      #pragma unroll
      for (int q = 0; q < SPLIT; ++q) rs += ws_recsum[((size_t)(b * SPLIT + q)) * Dd + d];
      s += rs * Wc[d * 2 + c];
    }
    out[b * 2 + c] = s * (1.f / (float)Ss) + bcv[c];
  }
  if (t == 0) {
    float s = 0.f;
    for (int i = 0; i < Bsz * SPLIT; ++i) s += ws_ae[i];
    out[3277059] = s / ((float)Bsz * (float)Ss * (float)Dd);
    float cs = 0.f;
    for (int kq = 0; kq < Kk; ++kq) cs += ws_topsum[kq];
    out[257] = -cs / ((float)Kk * (float)(Bsz / 4));
  }
}

extern "C" void kernel_launch(void* const* d_in, const int* in_sizes, int n_in,
                              void* d_out, int out_size, void* d_ws, size_t ws_size,
                              hipStream_t stream) {
  const float* f_input      = (const float*)d_in[0];
  const float* topic_vector = (const float*)d_in[2];
  const float* W1           = (const float*)d_in[3];
  const float* W2           = (const float*)d_in[4];
  const float* Wc           = (const float*)d_in[5];
  const float* bcv          = (const float*)d_in[6];
  float* out = (float*)d_out;
  float* ws  = (float*)d_ws;

  float* tvpP   = ws;                        // 128*64*2  = 16384
  float* W1pP   = tvpP + (Dd / 2) * KP * 2;  // 32*256*2  = 16384
  float* W2P    = W1pP + (KP / 2) * Hh * 2;  // 128*256*2 = 65536
  float* tpn    = W2P + (Hh / 2) * Dd * 2;   // 50*65536
  float* recsum = tpn + (size_t)Kk * NROW;   // 128*8*256
  float* aep    = recsum + Bsz * SPLIT * Dd; // 1024
  float* topsum = aep + Bsz * SPLIT;         // 50
  float* out_tpnn = out + 259;

  prep_kernel<<<1, 256, 0, stream>>>(topic_vector, W1, W2, tvpP, W1pP, W2P, out);
  main_kernel<<<Bsz * SPLIT / 2, 64, 0, stream>>>(f_input, tvpP, W1pP, W2P,
                                                  out_tpnn, tpn, recsum, aep);
  topk_kernel<<<Kk, 256, 0, stream>>>(tpn, topsum);
  final_kernel<<<1, 256, 0, stream>>>(Wc, bcv, recsum, aep, topsum, out);
}